// CTCLayer_61933428408324
// MI455X (gfx1250) — compile-verified
//
#include <hip/hip_runtime.h>
#include <hip/hip_bf16.h>
#include <cstdint>

// Problem constants (fixed by the reference setup_inputs).
#define TT 1024
#define CC 128            // num classes, blank = CC-1
#define LL 128            // label length
#define SS 257            // 2*LL + 1 lattice states
#define NEG_INFF (-1.0e30f)
#define EPSF (1.0e-7f)

typedef float v2f __attribute__((ext_vector_type(2)));
typedef float v8f __attribute__((ext_vector_type(8)));

#ifndef __has_builtin
#define __has_builtin(x) 0
#endif

#if __has_builtin(__builtin_amdgcn_wmma_f32_16x16x4_f32)
#define HAVE_WMMA4 1
#else
#define HAVE_WMMA4 0
#endif

#if __has_builtin(__builtin_amdgcn_global_load_async_to_lds_b32)
#define HAVE_ASYNC 1
#else
#define HAVE_ASYNC 0
#endif

typedef __attribute__((address_space(1))) int gint_t;
typedef __attribute__((address_space(3))) int lint_t;

__device__ __forceinline__ void async_load_f32(const float* g, float* l) {
#if HAVE_ASYNC
  // GLOBAL_LOAD_ASYNC_TO_LDS_B32: copies the float at global address g to LDS address l, tracked by ASYNCcnt.
  __builtin_amdgcn_global_load_async_to_lds_b32(
      (gint_t*)(float*)g, (lint_t*)l, 0, 0);
#else
  *l = *g;  // synchronous fallback through VGPRs
#endif
}

__device__ __forceinline__ void wait_async() {
#if HAVE_ASYNC
#if __has_builtin(__builtin_amdgcn_s_wait_asynccnt)
  __builtin_amdgcn_s_wait_asynccnt(0);
#else
  asm volatile("s_wait_asynccnt 0" ::: "memory");
#endif
#endif
}

// ---------------------------------------------------------------------------
// Kernel 1: logZ[b*T+t] = log( sum_c y_pred[b,t,c] + C*eps )
// One wave per 16-row tile; row sums via V_WMMA_F32_16X16X4_F32 with B = ones.
// A-matrix 16x4 f32 layout (ISA 7.12.2): lanes 0-15 hold K=0 (v0) / K=1 (v1),
// lanes 16-31 hold K=2,3 for the same M rows. With B all-ones, every column
// of D equals the row sum, so layout of B is irrelevant.
// ---------------------------------------------------------------------------
__global__ __launch_bounds__(256) void ctc_logz_kernel(
    const float* __restrict__ yp, float* __restrict__ logz, int nrows) {
  const int lane = threadIdx.x & 31;
  const int wid  = blockIdx.x * (blockDim.x >> 5) + (threadIdx.x >> 5);
  const int row0 = wid << 4;
  if (row0 >= nrows) return;           // wave-uniform exit, EXEC stays all-1s

  const int r = lane & 15;             // M row within tile
  const int h = lane >> 4;             // half: selects K pair {0,1} vs {2,3}
  const float* rowp = yp + (size_t)(row0 + r) * CC;

#if HAVE_WMMA4
  v8f acc = {0.f, 0.f, 0.f, 0.f, 0.f, 0.f, 0.f, 0.f};
  v2f ones; ones.x = 1.0f; ones.y = 1.0f;
#pragma unroll
  for (int c = 0; c < CC; c += 4) {
    const float* pp = rowp + c + 2 * h;
    v2f a; a.x = pp[0]; a.y = pp[1];
    // D = A(16x4) * ones(4x16) + C : accumulates row sums into every column.
    acc = __builtin_amdgcn_wmma_f32_16x16x4_f32(
        false, a, false, ones, (short)0, acc, false, false);
  }
  // D layout: lanes 0-15 vgpr v = D[v][lane]; lanes 16-31 vgpr v = D[v+8][lane-16]
  if (r == 0) {
    float* o = logz + row0 + h * 8;
#pragma unroll
    for (int v = 0; v < 8; ++v)
      o[v] = __logf(acc[v] + (float)CC * EPSF);
  }
#else
  float s = 0.f;
#pragma unroll
  for (int c = 0; c < CC; c += 4) {
    const float* pp = rowp + c + 2 * h;
    s += pp[0] + pp[1];
  }
  s += __shfl_xor(s, 16);              // combine the two column-halves
  if (h == 0) logz[row0 + r] = __logf(s + (float)CC * EPSF);
#endif
}

// ---------------------------------------------------------------------------
// Kernel 2: CTC forward recurrence. One block per batch element b.
// 288 threads (9 waves); thread s < 257 owns lattice state s, alpha in a
// register. Alpha is double-buffered in LDS (index parity t&1) so exactly one
// barrier per time step suffices:
//   writes to albuf[t&1] at step t vs reads of albuf[t&1] at step t+1
//   are separated by the end-of-step-t barrier; reads of albuf[(t-1)&1] at
//   step t vs its overwrite at step t+1 likewise.
// The 128-float emission row for step t+1 is async-prefetched into
// prow[(t+1)&1] while step t computes (its previous contents, step t-1's row,
// were last read before the end-of-step-(t-1) barrier).
// ---------------------------------------------------------------------------
__global__ __launch_bounds__(288) void ctc_alpha_kernel(
    const int* __restrict__ ytrue, const float* __restrict__ yp,
    const float* __restrict__ logz, float* __restrict__ out) {
  __shared__ float prow[2][CC];        // emission rows (raw p; +eps on use)
  __shared__ float albuf[2][SS + 3];   // alpha, stored at index s+2 (pad 2)

  const int tid = threadIdx.x;
  const int b   = blockIdx.x;
  const int s   = tid;
  const float* ypb = yp + (size_t)b * TT * CC;
  const float* lzb = logz + (size_t)b * TT;

  // Prefetch row t=0.
  if (tid < CC) async_load_f32(ypb + tid, &prow[0][tid]);
  if (tid < 2) { albuf[0][tid] = NEG_INFF; albuf[1][tid] = NEG_INFF; }

  // Per-state constants: ext label and skip-transition validity.
  // ext[s] = blank for even s, y_true[(s-1)/2] for odd s.
  // can_skip = odd s, s >= 3, and label != previous label.
  int extc = CC - 1;
  bool canskip = false;
  if ((s < SS) && (s & 1)) {
    const int li = s >> 1;
    extc = ytrue[(size_t)b * LL + li];
    if (li > 0) canskip = (extc != ytrue[(size_t)b * LL + li - 1]);
  }

  wait_async();
  __syncthreads();                      // prow[0] ready

  // t = 0 init: only states 0 and 1 reachable.
  float alpha = NEG_INFF;
  if (s < SS) {
    const float lp0 = __logf(prow[0][extc] + EPSF) - lzb[0];
    alpha = (s < 2) ? lp0 : NEG_INFF;
    albuf[0][s + 2] = alpha;
  }
  if (tid < CC) async_load_f32(ypb + (size_t)CC + tid, &prow[1][tid]);  // t=1
  wait_async();
  __syncthreads();                      // albuf[0] + prow[1] ready

  for (int t = 1; t < TT; ++t) {
    const int pc = t & 1;               // current parity
    const int pp = pc ^ 1;              // previous parity
    // Prefetch row t+1 into the buffer whose data (t-1) is no longer needed.
    if ((t + 1 < TT) && (tid < CC))
      async_load_f32(ypb + (size_t)(t + 1) * CC + tid, &prow[pp][tid]);

    const float lz = lzb[t];            // uniform -> scalar load
    if (s < SS) {
      const float pv     = prow[pc][extc];            // LDS gather
      const float a_prev = albuf[pp][s + 1];          // state s-1
      const float a_skip = canskip ? albuf[pp][s] : NEG_INFF;  // state s-2
      const float m   = fmaxf(alpha, fmaxf(a_prev, a_skip));
      const float lse = m + __logf(__expf(alpha - m) + __expf(a_prev - m) +
                                   __expf(a_skip - m));
      alpha = lse + (__logf(pv + EPSF) - lz);
      albuf[pc][s + 2] = alpha;
    }
    wait_async();
    __syncthreads();
  }

  // loss = -logaddexp(alpha_T[S-1], alpha_T[S-2])
  if (tid == 0) {
    const int lb = (TT - 1) & 1;
    const float aA = albuf[lb][SS + 1];  // state S-1 (final blank)
    const float aB = albuf[lb][SS];      // state S-2 (final label)
    const float m = fmaxf(aA, aB);
    out[b] = -(m + __logf(__expf(aA - m) + __expf(aB - m)));
  }
}

extern "C" void kernel_launch(void* const* d_in, const int* in_sizes, int n_in,
                              void* d_out, int out_size, void* d_ws, size_t ws_size,
                              hipStream_t stream) {
  const int*   ytrue = (const int*)d_in[0];    // [B, L] int32
  const float* ypred = (const float*)d_in[1];  // [B, T, C] float32
  float* out  = (float*)d_out;                 // [B, 1] float32
  float* logz = (float*)d_ws;                  // B*T floats scratch (512 KB)

  const int B = in_sizes[0] / LL;
  const int nrows = B * TT;
  const int tiles = nrows >> 4;                // 16 rows per wave
  const int blocks1 = (tiles + 7) / 8;         // 8 waves per 256-thread block

  ctc_logz_kernel<<<blocks1, 256, 0, stream>>>(ypred, logz, nrows);
  ctc_alpha_kernel<<<B, 288, 0, stream>>>(ytrue, ypred, logz, out);
}